// random_noiseM_39367670235562
// MI455X (gfx1250) — compile-verified
//
#include <hip/hip_runtime.h>
#include <hip/hip_bf16.h>

// ---------------------------------------------------------------------------
// BatchNorm1d (training stats over B*T per channel) + gaussian noise add.
//   B=16, T=512, D=8576  -> N = B*T = 8192 rows, D contiguous innermost.
// Memory-bound: ~1.12 GB traffic => ~48us floor @ 23.3 TB/s.
// Reduction pass uses V_WMMA_F32_16X16X4_F32 with an all-ones A matrix:
//   D[m][n] = sum_k B[k][n]  (+C accumulate)  -- 4 rows x 16 channels / wmma.
// Deterministic: per-chunk partials in workspace, fixed-order final reduce.
// ---------------------------------------------------------------------------

#define B_DIM 16
#define T_DIM 512
#define D_DIM 8576
#define N_ROWS (B_DIM * T_DIM)          // 8192
#define N_CHUNKS 32
#define ROWS_PER_CHUNK (N_ROWS / N_CHUNKS)  // 256
#define N_TILES (D_DIM / 16)            // 536 (D divides 16 exactly)
#define WAVES_PER_BLOCK 4
#define EPS 1e-5f
#define NOISE_STD 0.05f

typedef __attribute__((ext_vector_type(2))) float v2f;
typedef __attribute__((ext_vector_type(4))) float v4f;   // native vector: OK for nontemporal builtins
typedef __attribute__((ext_vector_type(8))) float v8f;

// ---------------------------------------------------------------------------
// Pass 1: per-channel partial sum / sum-of-squares via WMMA f32 16x16x4.
// Each wave owns one (16-channel tile, 256-row chunk). A = ones(16x4), so the
// accumulator's row 0 holds column sums regardless of the B-operand K layout.
// EXEC is all-ones for the whole wmma loop (no divergence before it).
// ---------------------------------------------------------------------------
__global__ __launch_bounds__(WAVES_PER_BLOCK * 32)
void bn_reduce_wmma(const float* __restrict__ x,
                    float* __restrict__ psum,   // [N_CHUNKS][D_DIM]
                    float* __restrict__ psq)    // [N_CHUNKS][D_DIM]
{
    const int wave_in_blk = threadIdx.x >> 5;
    const int lane        = threadIdx.x & 31;
    const int wid         = blockIdx.x * WAVES_PER_BLOCK + wave_in_blk;
    const int tile        = wid % N_TILES;
    const int chunk       = wid / N_TILES;

    const int ch      = tile * 16 + (lane & 15);
    const int rowHalf = (lane >> 4) * 2;            // lanes 0-15 -> rows {r,r+1}, 16-31 -> {r+2,r+3}

    const float* base = x + (size_t)chunk * ROWS_PER_CHUNK * D_DIM + ch;

    v2f ones; ones.x = 1.0f; ones.y = 1.0f;
    v8f accS = {};
    v8f accQ = {};

    for (int r = 0; r < ROWS_PER_CHUNK; r += 4) {
        const float* p = base + (size_t)(r + rowHalf) * D_DIM;
        v2f b;
        b.x = p[0];
        b.y = p[D_DIM];
        v2f bq;
        bq.x = b.x * b.x;
        bq.y = b.y * b.y;
        // 8 args: (neg_a, A, neg_b, B, c_mod, C, reuse_a, reuse_b)
        accS = __builtin_amdgcn_wmma_f32_16x16x4_f32(false, ones, false, b,
                                                     (short)0, accS, false, false);
        accQ = __builtin_amdgcn_wmma_f32_16x16x4_f32(false, ones, false, bq,
                                                     (short)0, accQ, false, false);
    }

    // C/D layout: VGPR0 = D[0][n] on lanes 0-15 (rows of D are identical since
    // A is all-ones). Lanes 0-15 publish the 16 channel partials. No atomics.
    if (lane < 16) {
        psum[(size_t)chunk * D_DIM + ch] = accS[0];
        psq [(size_t)chunk * D_DIM + ch] = accQ[0];
    }
}

// ---------------------------------------------------------------------------
// Pass 2: fold mu/var/gamma/beta into per-channel scale/shift.
//   scale = gamma * rsqrt(var + eps);  shift = beta - mu * scale
// Fixed-order chunk reduction => deterministic.
// ---------------------------------------------------------------------------
__global__ __launch_bounds__(256)
void bn_finalize(const float* __restrict__ psum,
                 const float* __restrict__ psq,
                 const float* __restrict__ gamma,
                 const float* __restrict__ beta,
                 float* __restrict__ scale,
                 float* __restrict__ shift)
{
    const int d = blockIdx.x * blockDim.x + threadIdx.x;
    if (d >= D_DIM) return;

    float s = 0.0f, q = 0.0f;
    for (int k = 0; k < N_CHUNKS; ++k) {
        s += psum[(size_t)k * D_DIM + d];
        q += psq [(size_t)k * D_DIM + d];
    }
    const float invN = 1.0f / (float)N_ROWS;
    const float mu   = s * invN;
    const float var  = fmaf(-mu, mu, q * invN);   // E[x^2] - mu^2 (biased)
    const float rstd = rsqrtf(var + EPS);
    const float sc   = rstd * gamma[d];
    scale[d] = sc;
    shift[d] = fmaf(-mu, sc, beta[d]);
}

// ---------------------------------------------------------------------------
// Pass 3: y = x*scale + shift + noise*0.05, float4-vectorized (b128 ops).
// Total float4 count 17,563,648 == 68608 * 256: no tail handling needed.
// noise is read once and y written once -> non-temporal; keep L2 for x and
// the hot scale/shift tables (34 KB each, resident in WGP$/L2).
// ---------------------------------------------------------------------------
__global__ __launch_bounds__(256)
void bn_apply(const v4f* __restrict__ x4,
              const v4f* __restrict__ n4,
              const v4f* __restrict__ scale4,  // [D_DIM/4]
              const v4f* __restrict__ shift4,  // [D_DIM/4]
              v4f* __restrict__ y4)
{
    const unsigned i  = blockIdx.x * 256u + threadIdx.x;
    const unsigned c4 = i % (D_DIM / 4);          // D_DIM % 4 == 0

    const v4f xv = x4[i];
    const v4f nv = __builtin_nontemporal_load(&n4[i]);
    const v4f sc = scale4[c4];
    const v4f sh = shift4[c4];

    v4f r;
    r.x = fmaf(xv.x, sc.x, fmaf(nv.x, NOISE_STD, sh.x));
    r.y = fmaf(xv.y, sc.y, fmaf(nv.y, NOISE_STD, sh.y));
    r.z = fmaf(xv.z, sc.z, fmaf(nv.z, NOISE_STD, sh.z));
    r.w = fmaf(xv.w, sc.w, fmaf(nv.w, NOISE_STD, sh.w));

    __builtin_nontemporal_store(r, &y4[i]);
}

// ---------------------------------------------------------------------------
extern "C" void kernel_launch(void* const* d_in, const int* in_sizes, int n_in,
                              void* d_out, int out_size, void* d_ws, size_t ws_size,
                              hipStream_t stream) {
    (void)in_sizes; (void)n_in; (void)out_size; (void)ws_size;

    const float* x     = (const float*)d_in[0];
    const float* noise = (const float*)d_in[1];
    const float* gamma = (const float*)d_in[2];
    const float* beta  = (const float*)d_in[3];
    float*       out   = (float*)d_out;

    // Workspace layout (floats): psum[32*D] | psq[32*D] | scale[D] | shift[D]
    float* ws    = (float*)d_ws;
    float* psum  = ws;
    float* psq   = ws + (size_t)N_CHUNKS * D_DIM;
    float* scale = ws + 2 * (size_t)N_CHUNKS * D_DIM;
    float* shift = scale + D_DIM;

    // Pass 1: 536 tiles * 32 chunks = 17152 waves, 4 waves per block.
    const int nWaves  = N_TILES * N_CHUNKS;
    const int nBlocks = nWaves / WAVES_PER_BLOCK;   // 4288 (divides exactly)
    bn_reduce_wmma<<<nBlocks, WAVES_PER_BLOCK * 32, 0, stream>>>(x, psum, psq);

    // Pass 2: tiny.
    bn_finalize<<<(D_DIM + 255) / 256, 256, 0, stream>>>(psum, psq, gamma, beta,
                                                         scale, shift);

    // Pass 3: elementwise fused normalize + noise.
    const unsigned n4 = (unsigned)((size_t)B_DIM * T_DIM * D_DIM / 4); // 17,563,648
    bn_apply<<<n4 / 256, 256, 0, stream>>>((const v4f*)x, (const v4f*)noise,
                                           (const v4f*)scale, (const v4f*)shift,
                                           (v4f*)out);
}